// Read_6932077216213
// MI455X (gfx1250) — compile-verified
//
#include <hip/hip_runtime.h>
#include <hip/hip_bf16.h>

// Problem constants (match reference)
#define B_    2048
#define N_    256
#define V_    256
#define C_    512
#define OUTC  262           // V + 1 + 1 + 3 + 1
#define EPSF  1e-16f
#define LROW  260           // LDS row stride in floats: 1040 B = 65*16 (B128-aligned), stride%64==4 banks

typedef float v2f __attribute__((ext_vector_type(2)));
typedef float v8f __attribute__((ext_vector_type(8)));

// ---------------------------------------------------------------------------
// Kernel A: out_ws[m][n] = sum_k E[m][k] * W[n][k] + b[n]   (2048 x 262)
// One wave per 16x16 tile, f32 WMMA 16x16x4, K-loop 512/4 = 128 WMMAs.
// ---------------------------------------------------------------------------
__global__ __launch_bounds__(32) void ntm_proj_wmma(
    const float* __restrict__ E, const float* __restrict__ W,
    const float* __restrict__ bias, float* __restrict__ outw)
{
    const int lane = threadIdx.x;
    const int n0 = blockIdx.x * 16;
    const int m0 = blockIdx.y * 16;
    const int hi = (lane >= 16) ? 1 : 0;
    const int l  = lane & 15;

    const int arow  = m0 + l;
    int bcol = n0 + l;
    if (bcol >= OUTC) bcol = OUTC - 1;          // clamp OOB loads; stores masked below

    const float* erow = E + (size_t)arow * C_;
    const float* wrow = W + (size_t)bcol * C_;

    v8f acc = {};
    for (int k = 0; k < C_; k += 4) {
        const int kb = k + 2 * hi;
        float2 ea = *(const float2*)(erow + kb);
        float2 wb = *(const float2*)(wrow + kb);
        v2f a, bb;
        a[0] = ea.x; a[1] = ea.y;
        bb[0] = wb.x; bb[1] = wb.y;
#if __has_builtin(__builtin_amdgcn_wmma_f32_16x16x4_f32)
        acc = __builtin_amdgcn_wmma_f32_16x16x4_f32(
            /*neg_a=*/false, a, /*neg_b=*/false, bb,
            /*c_mod=*/(short)0, acc, /*reuse_a=*/false, /*reuse_b=*/false);
#else
        acc[0] += a[0] * bb[0] + a[1] * bb[1];
#endif
    }

    const int ccol = n0 + l;
    if (ccol < OUTC) {
        const float bv = bias[ccol];
        #pragma unroll
        for (int j = 0; j < 8; ++j) {
            const int m = m0 + j + 8 * hi;
            outw[(size_t)m * OUTC + ccol] = acc[j] + bv;
        }
    }
}

// ---------------------------------------------------------------------------
// Kernel B: one 1024-thread block per batch b. M_b (256 KB) staged in LDS via
// async global->LDS DMA (NT hint: M is streamed once, keep it out of L2).
// Split-wait pipelining: rows 0..127 of pass 1 run while rows 128..255 land.
// ---------------------------------------------------------------------------
__global__ __launch_bounds__(1024) void ntm_head(
    const float* __restrict__ outw,    // (B, OUTC) projection
    const float* __restrict__ wprev,   // (B, N)
    const float* __restrict__ M,       // (B, N, V)
    float* __restrict__ outR,          // (B, V)
    float* __restrict__ outWt)         // (B, N)
{
    extern __shared__ float4 smem4[];           // M tile: 256 rows x LROW floats
    float* smem = (float*)smem4;

    __shared__ float s_kn[V_];
    __shared__ float s_sim[N_];
    __shared__ float s_w[N_];
    __shared__ float s_ra[1024];
    __shared__ float s_rb[1024];
    __shared__ float s_scal[8];   // 0:beta 1:g 2:s0 3:s1 4:s2 5:gamma 6:knorm

    const int tid = threadIdx.x;
    const int b   = blockIdx.x;
    const float* Mb = M + (size_t)b * N_ * V_;

    // ---- issue async global->LDS DMA of M_b (16384 chunks of 16B) ----
    // iteration it covers rows it*16 .. it*16+15 across the block; ASYNCcnt
    // completes in order, so "asynccnt<=8" == first 8 iters == rows 0..127 done.
    const unsigned ldsbase = (unsigned)(size_t)smem;   // low 32 bits of flat addr == LDS byte offset
    #pragma unroll
    for (int it = 0; it < 16; ++it) {
        const int idx = it * 1024 + tid;   // chunk id
        const int n = idx >> 6;            // row
        const int c = idx & 63;            // 16B chunk within row
        const unsigned laddr = ldsbase + (unsigned)(n * (LROW * 4) + c * 16);
        const float* g = Mb + n * V_ + c * 4;
        asm volatile("global_load_async_to_lds_b128 %0, %1, off th:TH_LOAD_NT"
                     :: "v"(laddr), "v"(g) : "memory");
    }

    // ---- overlap: head parameters from projection row ----
    float kraw = 0.f;
    if (tid < V_) { kraw = outw[(size_t)b * OUTC + tid]; s_ra[tid] = kraw * kraw; }
    __syncthreads();
    for (int s = 128; s > 0; s >>= 1) {
        if (tid < s) s_ra[tid] += s_ra[tid + s];
        __syncthreads();
    }
    if (tid == 0) {
        const float knorm = sqrtf(s_ra[0]);
        const float* h = outw + (size_t)b * OUTC + V_;
        const float xb = h[0], xg = h[1], x0 = h[2], x1 = h[3], x2 = h[4], xgm = h[5];
        const float beta  = fmaxf(xb, 0.f) + log1pf(expf(-fabsf(xb)));   // softplus
        const float g     = 1.f / (1.f + expf(-xg));                      // sigmoid
        const float mx = fmaxf(x0, fmaxf(x1, x2));
        const float e0 = expf(x0 - mx), e1 = expf(x1 - mx), e2 = expf(x2 - mx);
        const float es = e0 + e1 + e2;
        const float gamma = 1.f + fmaxf(xgm, 0.f) + log1pf(expf(-fabsf(xgm)));
        s_scal[0] = beta; s_scal[1] = g;
        s_scal[2] = e0 / es; s_scal[3] = e1 / es; s_scal[4] = e2 / es;
        s_scal[5] = gamma; s_scal[6] = knorm;
    }
    __syncthreads();
    if (tid < V_) s_kn[tid] = kraw / (s_scal[6] + EPSF);

    // ---- pass 1 (pipelined): sim_n = beta * (kn . M_n) / (||M_n|| + eps) ----
    // 4 lanes per row; per-part rotation (16*p1) does not put all 32 lanes on distinct banks.
    const int n1 = tid >> 2, p1 = tid & 3;

    // lower half of tile resident after first 8 DMA iterations
    asm volatile("s_wait_asynccnt 8" ::: "memory");
    __syncthreads();
    if (n1 < 128) {
        const float* row = smem + n1 * LROW;
        float dot = 0.f, sq = 0.f;
        #pragma unroll 4
        for (int j = 0; j < 64; ++j) {
            const int v = p1 * 64 + ((j + 16 * p1) & 63);
            const float m = row[v];
            dot += m * s_kn[v];
            sq  += m * m;
        }
        s_ra[tid] = dot; s_rb[tid] = sq;
    }

    // upper half after all DMA complete
    asm volatile("s_wait_asynccnt 0" ::: "memory");
    __syncthreads();
    if (n1 >= 128) {
        const float* row = smem + n1 * LROW;
        float dot = 0.f, sq = 0.f;
        #pragma unroll 4
        for (int j = 0; j < 64; ++j) {
            const int v = p1 * 64 + ((j + 16 * p1) & 63);
            const float m = row[v];
            dot += m * s_kn[v];
            sq  += m * m;
        }
        s_ra[tid] = dot; s_rb[tid] = sq;
    }
    __syncthreads();

    if (tid < N_) {
        const float dot = s_ra[4*tid] + s_ra[4*tid+1] + s_ra[4*tid+2] + s_ra[4*tid+3];
        const float sq  = s_rb[4*tid] + s_rb[4*tid+1] + s_rb[4*tid+2] + s_rb[4*tid+3];
        s_sim[tid] = s_scal[0] * dot / (sqrtf(sq) + EPSF);
    }
    __syncthreads();

    // ---- softmax over n ----
    if (tid < N_) s_ra[tid] = s_sim[tid];
    __syncthreads();
    for (int s = 128; s > 0; s >>= 1) {
        if (tid < s) s_ra[tid] = fmaxf(s_ra[tid], s_ra[tid + s]);
        __syncthreads();
    }
    const float smax = s_ra[0];
    __syncthreads();
    float e = 0.f;
    if (tid < N_) { e = expf(s_sim[tid] - smax); s_ra[tid] = e; }
    __syncthreads();
    for (int s = 128; s > 0; s >>= 1) {
        if (tid < s) s_ra[tid] += s_ra[tid + s];
        __syncthreads();
    }
    const float esum = s_ra[0];
    __syncthreads();

    // ---- interpolate with w_prev ----
    if (tid < N_) {
        const float wc = e / esum;
        const float g  = s_scal[1];
        s_w[tid] = g * wc + (1.f - g) * wprev[(size_t)b * N_ + tid];
    }
    __syncthreads();

    // ---- circular shift + sharpening ----
    float wp = 0.f;
    if (tid < N_) {
        const float wt = s_scal[2] * s_w[(tid + 1) & 255]
                       + s_scal[3] * s_w[tid]
                       + s_scal[4] * s_w[(tid + 255) & 255];
        wp = powf(wt, s_scal[5]);
        s_ra[tid] = wp;
    }
    __syncthreads();
    for (int s = 128; s > 0; s >>= 1) {
        if (tid < s) s_ra[tid] += s_ra[tid + s];
        __syncthreads();
    }
    const float psum = s_ra[0];
    __syncthreads();
    if (tid < N_) {
        const float wf = wp / (psum + EPSF);
        s_w[tid] = wf;
        outWt[(size_t)b * N_ + tid] = wf;
    }
    __syncthreads();

    // ---- pass 2: r_v = sum_n w_n * M[n][v], all 1024 threads ----
    // thread (v, p): n = 4j+p. Within a wave p is constant and v consecutive
    // -> LDS reads conflict-free; s_w[n] is a broadcast.
    {
        const int v = tid & 255, p = tid >> 8;
        float acc = 0.f;
        #pragma unroll 8
        for (int j = 0; j < 64; ++j) {
            const int n = 4 * j + p;
            acc += s_w[n] * smem[n * LROW + v];
        }
        s_ra[tid] = acc;
    }
    __syncthreads();
    if (tid < V_) {
        const float r = s_ra[tid] + s_ra[tid + 256] + s_ra[tid + 512] + s_ra[tid + 768];
        outR[(size_t)b * V_ + tid] = r;
    }
}

extern "C" void kernel_launch(void* const* d_in, const int* in_sizes, int n_in,
                              void* d_out, int out_size, void* d_ws, size_t ws_size,
                              hipStream_t stream) {
    const float* E     = (const float*)d_in[0];   // embeddings (B, C)
    const float* wprev = (const float*)d_in[1];   // (B, N)
    const float* W     = (const float*)d_in[2];   // (OUTC, C)
    const float* bias  = (const float*)d_in[3];   // (OUTC,)
    const float* M     = (const float*)d_in[4];   // (B, N, V)

    float* outR  = (float*)d_out;                 // (B, V)
    float* outWt = outR + (size_t)B_ * V_;        // (B, N)
    float* out_ws = (float*)d_ws;                 // (B, OUTC) projection scratch

    // Kernel A: projection GEMM via f32 WMMA
    dim3 gA((OUTC + 15) / 16, B_ / 16);
    ntm_proj_wmma<<<gA, 32, 0, stream>>>(E, W, bias, out_ws);

    // Kernel B: fused head, M tile LDS-resident (266,240 B dynamic + ~11 KB static < 320 KB)
    const size_t dynsz = (size_t)N_ * LROW * sizeof(float);
    hipFuncSetAttribute((const void*)ntm_head,
                        hipFuncAttributeMaxDynamicSharedMemorySize, (int)dynsz);
    ntm_head<<<B_, 1024, dynsz, stream>>>(out_ws, wprev, M, outR, outWt);
}